// MARCIA_65017214927368
// MI455X (gfx1250) — compile-verified
//
#include <hip/hip_runtime.h>
#include <hip/hip_bf16.h>

typedef float v2f __attribute__((ext_vector_type(2)));
typedef float v8f __attribute__((ext_vector_type(8)));

// ---------------- degree / normalization ----------------

__global__ void MARCIA_deg_init(float* __restrict__ deg, int n) {
  int i = blockIdx.x * blockDim.x + threadIdx.x;
  if (i < n) deg[i] = 1.0f;  // self-loop weight
}

__global__ void MARCIA_deg_acc(const int* __restrict__ dst, const float* __restrict__ ew,
                               float* __restrict__ deg, int E) {
  int e = blockIdx.x * blockDim.x + threadIdx.x;
  if (e < E) atomicAdd(&deg[dst[e]], ew[e]);
}

__global__ void MARCIA_dinv_k(float* __restrict__ deg, int n) {
  int i = blockIdx.x * blockDim.x + threadIdx.x;
  if (i < n) {
    float d = deg[i];
    deg[i] = d > 0.0f ? rsqrtf(d) : 0.0f;
  }
}

// ---------------- fp32 WMMA GEMM: H[n_rows x NDIM] = X[n_rows x 128] * W[128 x NDIM] --------
// One wave computes a 16x16 tile via 32 x V_WMMA_F32_16X16X4_F32 (K=128).
// B fragments for the wave's column tile are preloaded once and reused over 4 row slabs.

template <int NDIM>
__global__ void MARCIA_gemm_wmma(const float* __restrict__ X, const float* __restrict__ W,
                                 float* __restrict__ H, int n_rows) {
  constexpr int CT = NDIM / 16;                    // column tiles
  const int lane = threadIdx.x & 31;
  const int wave = threadIdx.x >> 5;
  const int gw   = blockIdx.x * (blockDim.x >> 5) + wave;
  const int ct   = gw % CT;                        // column tile id
  const int rb   = gw / CT;                        // 64-row block id
  const int hi   = lane >> 4;                      // 0: lanes 0-15, 1: lanes 16-31
  const int lo   = lane & 15;
  const int col  = ct * 16 + lo;

  // B (4x16 f32): VGPR0 -> K = k+2*hi, VGPR1 -> K = k+2*hi+1, N = lo
  v2f bfrag[32];
#pragma unroll
  for (int kk = 0; kk < 32; ++kk) {
    const int k = kk * 4 + 2 * hi;
    bfrag[kk].x = W[(size_t)(k + 0) * NDIM + col];
    bfrag[kk].y = W[(size_t)(k + 1) * NDIM + col];
  }

#pragma unroll
  for (int s = 0; s < 4; ++s) {
    const int row0 = rb * 64 + s * 16;
    if (row0 >= n_rows) break;                     // wave-uniform guard
    // A (16x4 f32): lane holds A[row0+lo][k+2*hi], A[row0+lo][k+2*hi+1]
    const float* xrow = X + (size_t)(row0 + lo) * 128 + 2 * hi;
    v8f c = {};
#pragma unroll
    for (int kk = 0; kk < 32; ++kk) {
      v2f a = *(const v2f*)(xrow + kk * 4);        // 8-byte aligned
      c = __builtin_amdgcn_wmma_f32_16x16x4_f32(
          /*neg_a=*/false, a, /*neg_b=*/false, bfrag[kk],
          /*c_mod=*/(short)0, c, /*reuse_a=*/false, /*reuse_b=*/false);
    }
    // D (16x16 f32): VGPR g -> M = g + 8*hi, N = lo
    float* hrow = H + (size_t)row0 * NDIM + col;
#pragma unroll
    for (int g = 0; g < 8; ++g) hrow[(size_t)(g + 8 * hi) * NDIM] = c[g];
  }
}

// ---------------- graph aggregation ----------------

// out[v] = h[v] * dinv[v]^2  (self-loop term; also initializes the accumulator)
template <int F>
__global__ void MARCIA_selfloop(const float* __restrict__ H, const float* __restrict__ dinv,
                                float* __restrict__ out, int n) {
  size_t i = (size_t)blockIdx.x * blockDim.x + threadIdx.x;
  if (i < (size_t)n * F) {
    int node = (int)(i / F);
    float di = dinv[node];
    out[i] = H[i] * di * di;
  }
}

// out[dst] += h[src] * (dinv[src]*ew*dinv[dst]) ; one wave per edge
template <int F>
__global__ void MARCIA_scatter(const int* __restrict__ src, const int* __restrict__ dst,
                               const float* __restrict__ ew, const float* __restrict__ dinv,
                               const float* __restrict__ H, float* __restrict__ out, int E) {
  const int lane = threadIdx.x & 31;
  const int wave = threadIdx.x >> 5;
  int e = blockIdx.x * (blockDim.x >> 5) + wave;
  const int stride = gridDim.x * (blockDim.x >> 5);
  for (; e < E; e += stride) {
    const int s = src[e];
    const int d = dst[e];
    const float norm = dinv[s] * ew[e] * dinv[d];
    const float* hs = H + (size_t)s * F;
    float* od = out + (size_t)d * F;
    for (int f = lane; f < F; f += 32)
      atomicAdd(&od[f], hs[f] * norm);
  }
}

template <int F>
__global__ void MARCIA_bias_relu(const float* __restrict__ in, const float* __restrict__ bias,
                                 float* __restrict__ out, int n) {
  size_t i = (size_t)blockIdx.x * blockDim.x + threadIdx.x;
  if (i < (size_t)n * F) {
    float v = in[i] + bias[i & (F - 1)];
    out[i] = v > 0.0f ? v : 0.0f;
  }
}

// ---------------- launcher ----------------

extern "C" void kernel_launch(void* const* d_in, const int* in_sizes, int n_in,
                              void* d_out, int out_size, void* d_ws, size_t ws_size,
                              hipStream_t stream) {
  const float* x  = (const float*)d_in[0];
  const int*   ei = (const int*)d_in[1];
  const float* ew = (const float*)d_in[2];
  const float* W1 = (const float*)d_in[3];
  const float* b1 = (const float*)d_in[4];
  const float* W2 = (const float*)d_in[5];
  const float* b2 = (const float*)d_in[6];

  const int n = in_sizes[0] / 128;   // 50000
  const int E = in_sizes[2];         // 500000
  const int* src = ei;
  const int* dst = ei + E;
  float* out = (float*)d_out;

  // workspace: dinv | bufA (n x 128) | bufB (n x 128)
  float* ws   = (float*)d_ws;
  size_t Npad = ((size_t)n + 255) & ~(size_t)255;
  float* dinv = ws;
  float* bufA = dinv + Npad;
  float* bufB = bufA + (size_t)n * 128;

  const int TPB = 256;  // 8 waves per block (wave32)
  const int nb_n = (n + TPB - 1) / TPB;
  const int nb_e = (E + TPB - 1) / TPB;

  // symmetric normalization factors
  MARCIA_deg_init<<<nb_n, TPB, 0, stream>>>(dinv, n);
  MARCIA_deg_acc<<<nb_e, TPB, 0, stream>>>(dst, ew, dinv, E);
  MARCIA_dinv_k<<<nb_n, TPB, 0, stream>>>(dinv, n);

  const int slabs   = (n + 15) / 16;        // 3125
  const int rblocks = (slabs + 3) / 4;      // 782

  // ---- layer 1 ----
  {
    int waves = rblocks * (128 / 16);
    MARCIA_gemm_wmma<128><<<(waves + 7) / 8, TPB, 0, stream>>>(x, W1, bufA, n);
  }
  {
    int nb = (int)(((size_t)n * 128 + TPB - 1) / TPB);
    MARCIA_selfloop<128><<<nb, TPB, 0, stream>>>(bufA, dinv, bufB, n);
    MARCIA_scatter<128><<<2048, TPB, 0, stream>>>(src, dst, ew, dinv, bufA, bufB, E);
    MARCIA_bias_relu<128><<<nb, TPB, 0, stream>>>(bufB, b1, bufA, n);  // a1 -> bufA
  }

  // ---- layer 2 ----
  {
    int waves = rblocks * (64 / 16);
    MARCIA_gemm_wmma<64><<<(waves + 7) / 8, TPB, 0, stream>>>(bufA, W2, bufB, n);
  }
  {
    int nb = (int)(((size_t)n * 64 + TPB - 1) / TPB);
    MARCIA_selfloop<64><<<nb, TPB, 0, stream>>>(bufB, dinv, out, n);
    MARCIA_scatter<64><<<2048, TPB, 0, stream>>>(src, dst, ew, dinv, bufB, out, E);
    MARCIA_bias_relu<64><<<nb, TPB, 0, stream>>>(out, b2, out, n);
  }
}